// TimeLayerNorm_48850958024788
// MI455X (gfx1250) — compile-verified
//
#include <hip/hip_runtime.h>

typedef __attribute__((ext_vector_type(2))) float v2f;
typedef __attribute__((ext_vector_type(8))) float v8f;

#define TLN_S 4096
#define TLN_B 8
#define TLN_C 1024
#define TLN_NC (TLN_B * TLN_C)          /* 8192 channels */
#define TLN_CHUNK 128
#define TLN_NCHUNK (TLN_S / TLN_CHUNK)  /* 32 chunks */
#define TLN_EPS 1e-5f

// ---------------------------------------------------------------------------
// Pass 1: per-chunk partial sums of x and x^2 for every channel.
// tid -> (chunk q, channel ch); consecutive threads hit consecutive channels,
// so every step is a fully coalesced 1 KB block load.
// ---------------------------------------------------------------------------
extern "C" __global__ __launch_bounds__(256)
void tln_partials(const float* __restrict__ x,
                  float* __restrict__ psum,
                  float* __restrict__ pssum) {
  const int tid = blockIdx.x * 256 + threadIdx.x;
  const int ch  = tid % TLN_NC;
  const int q   = tid / TLN_NC;
  const float* xp = x + (size_t)q * TLN_CHUNK * TLN_NC + ch;
  float s = 0.0f, ss = 0.0f;
#pragma unroll 8
  for (int i = 0; i < TLN_CHUNK; ++i) {
    const float v = xp[(size_t)i * TLN_NC];
    s  += v;
    ss += v * v;
  }
  psum[q * TLN_NC + ch]  = s;
  pssum[q * TLN_NC + ch] = ss;
}

// ---------------------------------------------------------------------------
// Pass 2: each wave owns (chunk q, 16-channel group). Lookback over previous
// chunks' partials (L2-resident, 2 MiB total), then scans its 128 rows in
// 16x16 tiles using v_wmma_f32_16x16x4_f32 with a lower-triangular ones
// A-matrix: prefix = L * X. Carries broadcast with a wave32 shuffle from the
// row-15 accumulator lane.
//
// C/D layout (ISA 7.12.2): lane l holds column j = l&15, VGPR r holds row
// r + 8*(l>>4). A(16x4 f32): VGPR0 = K{0|2}, VGPR1 = K{1|3} by half-wave.
// B(4x16 f32): VGPR0 = K0(lo)/K2(hi), VGPR1 = K1(lo)/K3(hi), lane%16 = N.
//
// Normalization uses v_rcp_f32 / v_rsq_f32 (TRANS ops, co-execute with VALU)
// instead of the IEEE divide fixup chains: pos is a benign positive integer
// and the reference itself uses lax.rsqrt, so ~1ulp approx error is fine and
// this keeps the kernel on the memory roofline instead of going VALU-bound.
// ---------------------------------------------------------------------------
extern "C" __global__ __launch_bounds__(256)
void tln_scan(const float* __restrict__ x,
              const float* __restrict__ pe,
              const float* __restrict__ weight,
              const float* __restrict__ bias,
              const float* __restrict__ psum,
              const float* __restrict__ pssum,
              float* __restrict__ out) {
  constexpr int NGB = TLN_NC / (16 * 8);  // 64 channel-group blocks
  const int gb   = blockIdx.x % NGB;
  const int q    = blockIdx.x / NGB;
  const int wave = threadIdx.x >> 5;
  const int lane = threadIdx.x & 31;
  const int j    = lane & 15;             // column within 16-channel group
  const int hi   = lane >> 4;             // half-wave select
  const int ch   = (gb * 8 + wave) * 16 + j;
  const int c    = ch & (TLN_C - 1);

  // Per-channel constants + pseudo-embedding seed.
  const float w  = weight[c] + 1.0f;
  const float bb = bias[c];
  const float p0 = pe[c];
  const float p1 = pe[TLN_C + c];
  float carry_s = p0 + p1;
  float carry_q = p0 * p0 + p1 * p1;

  // Lookback: accumulate totals of all preceding chunks for this channel.
  for (int t = 0; t < q; ++t) {
    carry_s += psum[t * TLN_NC + ch];
    carry_q += pssum[t * TLN_NC + ch];
  }

  const int s0 = q * TLN_CHUNK;
  for (int t16 = 0; t16 < TLN_CHUNK / 16; ++t16) {
    const int sb = s0 + t16 * 16;
    const float* xt = x + (size_t)sb * TLN_NC + ch;

    // x tile in C/D layout (for the final elementwise math + subtraction).
    float xr[8];
#pragma unroll
    for (int r = 0; r < 8; ++r)
      xr[r] = xt[(size_t)(r + 8 * hi) * TLN_NC];

    v8f acc = {0.f, 0.f, 0.f, 0.f, 0.f, 0.f, 0.f, 0.f};  // prefix(x)
    v8f acq = {0.f, 0.f, 0.f, 0.f, 0.f, 0.f, 0.f, 0.f};  // prefix(x^2)
#pragma unroll
    for (int p = 0; p < 4; ++p) {
      const int k0 = 4 * p + 2 * hi;         // K of first operand VGPR
      v2f A;                                  // lower-triangular ones: L[M][k], M = j
      A.x = (k0     <= j) ? 1.0f : 0.0f;
      A.y = (k0 + 1 <= j) ? 1.0f : 0.0f;
      const float b0 = xt[(size_t)k0 * TLN_NC];        // L0-cache hits (tile is 1 KB)
      const float b1 = xt[(size_t)(k0 + 1) * TLN_NC];
      v2f Bv; Bv.x = b0;      Bv.y = b1;
      v2f Bq; Bq.x = b0 * b0; Bq.y = b1 * b1;
      acc = __builtin_amdgcn_wmma_f32_16x16x4_f32(false, A, false, Bv, (short)0, acc, false, false);
      acq = __builtin_amdgcn_wmma_f32_16x16x4_f32(false, A, false, Bq, (short)0, acq, false, false);
    }

    // Tile totals = inclusive prefix at row 15 = VGPR 7 of lanes 16..31.
    const float tot_s = __shfl(acc[7], 16 + j, 32);
    const float tot_q = __shfl(acq[7], 16 + j, 32);

    float* ot = out + (size_t)sb * TLN_NC + ch;
#pragma unroll
    for (int r = 0; r < 8; ++r) {
      const int row = r + 8 * hi;
      const float sum    = carry_s + acc[r];
      const float ssum   = carry_q + acq[r];
      const float rpos   = __builtin_amdgcn_rcpf(3.0f + (float)(sb + row));
      const float mean   = sum * rpos;
      const float var    = ssum * rpos - mean * mean;
      const float invstd = __builtin_amdgcn_rsqf(var + TLN_EPS);
      ot[(size_t)row * TLN_NC] = (xr[r] - mean) * (w * invstd) + bb;
    }
    carry_s += tot_s;
    carry_q += tot_q;
  }
}

extern "C" void kernel_launch(void* const* d_in, const int* in_sizes, int n_in,
                              void* d_out, int out_size, void* d_ws, size_t ws_size,
                              hipStream_t stream) {
  const float* x      = (const float*)d_in[0];  // (4096, 8, 1024) f32
  const float* pe     = (const float*)d_in[1];  // (2, 1024) f32
  const float* weight = (const float*)d_in[2];  // (1024,) f32
  const float* bias   = (const float*)d_in[3];  // (1024,) f32
  float* out   = (float*)d_out;
  float* psum  = (float*)d_ws;                          // 32*8192 f32 = 1 MiB
  float* pssum = psum + (size_t)TLN_NCHUNK * TLN_NC;    // second 1 MiB

  dim3 blk(256);
  dim3 g1((TLN_NCHUNK * TLN_NC) / 256);                 // 1024 blocks
  tln_partials<<<g1, blk, 0, stream>>>(x, psum, pssum);

  constexpr int NGB = TLN_NC / (16 * 8);
  dim3 g2(NGB * TLN_NCHUNK);                            // 2048 blocks, 8 waves each
  tln_scan<<<g2, blk, 0, stream>>>(x, pe, weight, bias, psum, pssum, out);
}